// GCMConv_75471165325591
// MI455X (gfx1250) — compile-verified
//
#include <hip/hip_runtime.h>
#include <cstdint>

typedef __attribute__((ext_vector_type(2))) float v2f;
typedef __attribute__((ext_vector_type(8))) float v8f;
typedef __attribute__((ext_vector_type(4))) unsigned int u32x4;
typedef __attribute__((ext_vector_type(4))) int i32x4;
typedef __attribute__((ext_vector_type(8))) int i32x8;

#define NSITES   16384
#define S_PER_WG 8
#define NCOLS    (S_PER_WG * 18)   /* 144 columns: 8 sites x 18 reals  */
#define TSTRIDE  (NCOLS + 2)       /* LDS row stride pad (bank spread) */
#define THREADS  128

// DIMS = (8,8,16,16); site = ((d0*8 + d1)*16 + d2)*16 + d3
__device__ __forceinline__ int neighbor_site(int s, int axis) {
  switch (axis) {
    case 0: { int d = (s >> 11) & 7;  return (s & ~(7  << 11)) | (((d + 1) & 7)  << 11); }
    case 1: { int d = (s >> 8)  & 7;  return (s & ~(7  << 8))  | (((d + 1) & 7)  << 8);  }
    case 2: { int d = (s >> 4)  & 15; return (s & ~(15 << 4))  | (((d + 1) & 15) << 4);  }
    default:{ int d =  s        & 15; return (s & ~15)         | ((d + 1) & 15);         }
  }
}

// Fused: TDM-staged x tile + transport + t_w/w_full assembly +
// (weight x T) GEMM via WMMA f32 16x16x4 + final small complex matmuls.
// Complex 3x3 layout in 18 floats: e = (i*3+j)*2 + c.
__global__ __launch_bounds__(THREADS)
void gcmconv_kernel(const float* __restrict__ x,      // (16384, 8, 3, 3, 2)
                    const float* __restrict__ weight, // (4, 9, 33)
                    float* __restrict__ out)          // (16384, 8, 3, 3, 2)
{
  const int tid      = threadIdx.x;
  const int lane     = tid & 31;
  const int wv       = tid >> 5;
  const int siteBase = blockIdx.x * S_PER_WG;

  __shared__ float xlds[S_PER_WG][144];     // this WG's x tile (8 ch x 18)
  __shared__ float Wmat[48][36];            // weight padded 36x33 -> 48x36
  __shared__ float Tlds[36][TSTRIDE];       // t_w_full rows (K) x (site*18+e)
  __shared__ float Mlds[37][TSTRIDE];       // GEMM result; row 36 = dummy spill
  __shared__ float Wf[S_PER_WG][9][18];     // w_full per site

  // One transport task per thread: (site, axis, field)
  const int s_tr     = tid >> 4;        // 0..7
  const int axis_tr  = (tid >> 2) & 3;  // 0..3
  const int field_tr = tid & 3;         // 0..3
  const int nb_tr    = neighbor_site(siteBase + s_tr, axis_tr);
  const float* wn_ptr = &x[((size_t)nb_tr * 8 + 4 + field_tr) * 18];
  __builtin_prefetch(wn_ptr, 0, 0);     // global_prefetch_b8: neighbor w row

  // ---- Issue TDM: DMA this WG's contiguous x tile into LDS ---------------
  if (wv == 0) {
    const unsigned lds_off = (unsigned)(uintptr_t)&xlds[0][0]; // flat[31:0] == LDS offset
    const unsigned long long ga =
        (unsigned long long)(uintptr_t)(x + (size_t)siteBase * 144);
    u32x4 g0;
    g0.x = 1u;                                    // count=1 (valid user D#)
    g0.y = lds_off;                               // lds_addr
    g0.z = (unsigned)(ga & 0xffffffffull);        // global_addr[31:0]
    g0.w = (unsigned)((ga >> 32) & 0x01ffffffull) // global_addr[56:32]
         | (2u << 30);                            // type = 2 ("image")
    i32x8 g1;
    g1[0] = (2 << 16);    // data_size = 4B; no multicast/pad/iterate
    g1[1] = (144 << 16);  // tensor_dim0[15:0] = 144
    g1[2] = (8 << 16);    // tensor_dim0[31:16]=0 | tensor_dim1[15:0] = 8
    g1[3] = (144 << 16);  // tensor_dim1[31:16]=0 | tile_dim0 = 144
    g1[4] = 8;            // tile_dim1 = 8, tile_dim2 = 0
    g1[5] = 144;          // tensor_dim0_stride[31:0] = 144
    g1[6] = 0;            // stride0[47:32] | stride1[15:0]
    g1[7] = 0;            // stride1[47:16]
    i32x4 gz = {0, 0, 0, 0};
#if __has_include(<hip/amd_detail/amd_gfx1250_TDM.h>)
    i32x8 gz8 = {0, 0, 0, 0, 0, 0, 0, 0};
    __builtin_amdgcn_tensor_load_to_lds(g0, g1, gz, gz, gz8, 0);
#else
    __builtin_amdgcn_tensor_load_to_lds(g0, g1, gz, gz, 0);
#endif
  }

  // ---- Phase 0 (overlaps TDM): stage padded weight matrix ----------------
  for (int idx = tid; idx < 48 * 36; idx += THREADS) {
    int r = idx / 36, c = idx % 36;
    Wmat[r][c] = (r < 36 && c < 33) ? weight[r * 33 + c] : 0.0f;
  }
  // zero K-pad rows 33..35 of T
  for (int idx = tid; idx < 3 * TSTRIDE; idx += THREADS)
    Tlds[33 + idx / TSTRIDE][idx % TSTRIDE] = 0.0f;
  // identity row 32 of T (unit channel of t_w_full)
  for (int idx = tid; idx < S_PER_WG * 18; idx += THREADS) {
    int s = idx / 18, e = idx % 18;
    int c = e & 1, jj = (e >> 1) % 3, ii = (e >> 1) / 3;
    Tlds[32][s * 18 + e] = (ii == jj && c == 0) ? 1.0f : 0.0f;
  }

  if (wv == 0) __builtin_amdgcn_s_wait_tensorcnt(0);  // TDM tile landed
  __syncthreads();

  // ---- Phase 1a: w_full = [w, w^dagger, I]  (site-local: from LDS tile) --
  for (int idx = tid; idx < S_PER_WG * 9 * 18; idx += THREADS) {
    int s = idx / 162, r = idx % 162, v = r / 18, e = r % 18;
    int c = e & 1, jj = (e >> 1) % 3, ii = (e >> 1) / 3;
    float val;
    if (v < 4) {
      val = xlds[s][(4 + v) * 18 + (ii * 3 + jj) * 2 + c];
    } else if (v < 8) {              // dagger of x channel v (v in 4..7)
      float t = xlds[s][v * 18 + (jj * 3 + ii) * 2 + c];
      val = c ? -t : t;
    } else {                         // identity
      val = (ii == jj && c == 0) ? 1.0f : 0.0f;
    }
    Wf[s][v][e] = val;
  }

  // ---- Phase 1b: transports -> T rows 0..15 and daggers 16..31 -----------
  {
    float U[18], Wn[18], T1[18], wt[18];
    const float* up = &xlds[s_tr][axis_tr * 18];
#pragma unroll
    for (int e = 0; e < 18; ++e) { U[e] = up[e]; Wn[e] = wn_ptr[e]; }

    // T1 = U * Wn  (complex 3x3)
#pragma unroll
    for (int i = 0; i < 3; ++i)
#pragma unroll
      for (int k = 0; k < 3; ++k) {
        float tr = 0.f, ti = 0.f;
#pragma unroll
        for (int j = 0; j < 3; ++j) {
          float ar = U[(i*3+j)*2], ai = U[(i*3+j)*2+1];
          float br = Wn[(j*3+k)*2], bi = Wn[(j*3+k)*2+1];
          tr += ar * br - ai * bi;
          ti += ar * bi + ai * br;
        }
        T1[(i*3+k)*2] = tr; T1[(i*3+k)*2+1] = ti;
      }
    // wt = T1 * U^H : wt[i][k] = sum_j T1[i][j] * conj(U[k][j])
#pragma unroll
    for (int i = 0; i < 3; ++i)
#pragma unroll
      for (int k = 0; k < 3; ++k) {
        float tr = 0.f, ti = 0.f;
#pragma unroll
        for (int j = 0; j < 3; ++j) {
          float ar = T1[(i*3+j)*2], ai = T1[(i*3+j)*2+1];
          float br = U[(k*3+j)*2],  bi = U[(k*3+j)*2+1];
          tr += ar * br + ai * bi;
          ti += ai * br - ar * bi;
        }
        wt[(i*3+k)*2] = tr; wt[(i*3+k)*2+1] = ti;
      }

    const int ch = axis_tr * 4 + field_tr;
#pragma unroll
    for (int i = 0; i < 3; ++i)
#pragma unroll
      for (int k = 0; k < 3; ++k) {
        float re = wt[(i*3+k)*2], im = wt[(i*3+k)*2+1];
        Tlds[ch][s_tr*18 + (i*3+k)*2]          = re;
        Tlds[ch][s_tr*18 + (i*3+k)*2 + 1]      = im;
        Tlds[16 + ch][s_tr*18 + (k*3+i)*2]     = re;   // dagger: transpose...
        Tlds[16 + ch][s_tr*18 + (k*3+i)*2 + 1] = -im;  // ...and conjugate
      }
  }

  __syncthreads();

  // ---- Phase 2: M(36x144) = Wmat(48x36-pad) x T(36x144), WMMA f32 16x16x4
  {
    const int m16 = lane & 15;
    const int kb  = (lane >> 4) * 2;        // ISA 7.12.2: lanes 16-31 hold K+2
    for (int t = wv; t < 27; t += 4) {      // 3 M-tiles x 9 N-tiles, wave-uniform
      const int mt = t / 9, nt = t % 9;
      v8f acc = {};
#pragma unroll
      for (int k0 = 0; k0 < 36; k0 += 4) {
        v2f a, b;
        a.x = Wmat[mt * 16 + m16][k0 + kb];
        a.y = Wmat[mt * 16 + m16][k0 + kb + 1];
        b.x = Tlds[k0 + kb][nt * 16 + m16];
        b.y = Tlds[k0 + kb + 1][nt * 16 + m16];
        acc = __builtin_amdgcn_wmma_f32_16x16x4_f32(
            false, a, false, b, (short)0, acc, false, false);
      }
      const int rowBase = mt * 16 + (lane >> 4) * 8;  // C/D: vgpr r -> row r / r+8
      const int col     = nt * 16 + m16;
#pragma unroll
      for (int r = 0; r < 8; ++r) {
        int row = rowBase + r;
        row = (row < 36) ? row : 36;   // clamp pad rows to dummy spill row
        Mlds[row][col] = acc[r];       // unconditional store: no exec churn
      }
    }
  }

  __syncthreads();

  // ---- Phase 3: w_out[u] = sum_v w_full[v] * M[u*9+v]  (complex 3x3) ------
  for (int idx = tid; idx < S_PER_WG * 36; idx += THREADS) {
    int s = idx / 36, rem = idx % 36, u = rem / 9, e = rem % 9;
    int i = e / 3, k = e % 3;
    float cr = 0.f, ci = 0.f;
#pragma unroll
    for (int v = 0; v < 9; ++v) {
      const float* wf = &Wf[s][v][0];
      const float* mr = &Mlds[u * 9 + v][s * 18];
#pragma unroll
      for (int j = 0; j < 3; ++j) {
        float ar = wf[(i*3+j)*2], ai = wf[(i*3+j)*2+1];
        float br = mr[(j*3+k)*2], bi = mr[(j*3+k)*2+1];
        cr += ar * br - ai * bi;
        ci += ar * bi + ai * br;
      }
    }
    const int sG = siteBase + s;
    out[((sG * 8 + 4 + u) * 9 + e) * 2 + 0] = cr;
    out[((sG * 8 + 4 + u) * 9 + e) * 2 + 1] = ci;
  }

  // ---- Phase 4: pass through gauge links u (channels 0..3, from LDS tile) -
  for (int idx = tid; idx < S_PER_WG * 72; idx += THREADS) {
    int s = idx / 72, off = idx % 72;
    out[(size_t)(siteBase + s) * 144 + off] = xlds[s][off];
  }
}

extern "C" void kernel_launch(void* const* d_in, const int* in_sizes, int n_in,
                              void* d_out, int out_size, void* d_ws, size_t ws_size,
                              hipStream_t stream) {
  (void)in_sizes; (void)n_in; (void)d_ws; (void)ws_size; (void)out_size;
  const float* x      = (const float*)d_in[0];
  const float* weight = (const float*)d_in[1];
  float* out          = (float*)d_out;
  gcmconv_kernel<<<dim3(NSITES / S_PER_WG), dim3(THREADS), 0, stream>>>(x, weight, out);
}